// Raymarcher_10539849744786
// MI455X (gfx1250) — compile-verified
//
#include <hip/hip_runtime.h>
#include <hip/hip_bf16.h>
#include <stdint.h>

// NeRF-style ray compositor for MI455X (gfx1250, wave32).
// Memory-bound (204 MB / ~0.13 GFLOP) -> optimize the data-movement path:
// async global->LDS DMA (ASYNCcnt) with double buffering, wave-per-ray scan.

#define N_SAMPLES        128
#define WAVES_PER_BLOCK  8
#define BLOCK_SIZE       256
#define RAY_FLOATS       768   // 128 sigma + 128 dists + 128 z + 384 rgb
#define RAYS_PER_WAVE    4

#if __has_builtin(__builtin_amdgcn_global_load_async_to_lds_b128)
#define HAVE_ASYNC_B128 1
#endif
#if __has_builtin(__builtin_amdgcn_s_wait_asynccnt)
#define HAVE_WAIT_ASYNC 1
#endif

// Builtin signature (from clang diagnostic): param1 is
//   __attribute__((vector_size(16))) int  addrspace(1)*   (global int4*)
// so mirror that exact vector type for both pointer args.
typedef int v4i __attribute__((vector_size(16)));
typedef __attribute__((address_space(1))) void* as1_void;
typedef __attribute__((address_space(3))) void* as3_void;
typedef __attribute__((address_space(1))) v4i*  as1_v4i;
typedef __attribute__((address_space(3))) v4i*  as3_v4i;

__device__ __forceinline__ void async_b128(const float* g, float* l) {
#ifdef HAVE_ASYNC_B128
  __builtin_amdgcn_global_load_async_to_lds_b128(
      (as1_v4i)(as1_void)g, (as3_v4i)(as3_void)l, 0, 0);
#else
  unsigned loff = (unsigned)(uintptr_t)l;            // low 32 bits == LDS offset
  unsigned long long ga = (unsigned long long)(uintptr_t)g;
  asm volatile("global_load_async_to_lds_b128 %0, %1, off"
               :: "v"(loff), "v"(ga) : "memory");
#endif
}

template <int N>
__device__ __forceinline__ void wait_asynccnt() {
#ifdef HAVE_WAIT_ASYNC
  __builtin_amdgcn_s_wait_asynccnt(N);
#else
  asm volatile("s_wait_asynccnt %0" :: "n"(N) : "memory");
#endif
  asm volatile("" ::: "memory");                     // fence LDS reads below the wait
}

__global__ __launch_bounds__(BLOCK_SIZE) void
raymarch_composite_kernel(const float* __restrict__ sigma,
                          const float* __restrict__ rgb,
                          const float* __restrict__ dists,
                          const float* __restrict__ zvals,
                          const float* __restrict__ bg,
                          float* __restrict__ out,
                          int n_rays)
{
  __shared__ __align__(16) float smem[2 * WAVES_PER_BLOCK * RAY_FLOATS]; // 48 KB

  const int lane    = threadIdx.x & 31;
  const int wib     = threadIdx.x >> 5;
  const int wave    = (blockIdx.x * BLOCK_SIZE + threadIdx.x) >> 5;
  const int n_waves = (gridDim.x * BLOCK_SIZE) >> 5;

  float* buf0 = smem + (size_t)wib * RAY_FLOATS;
  float* buf1 = smem + (size_t)(WAVES_PER_BLOCK + wib) * RAY_FLOATS;

  // 6 async b128 DMA issues per ray: all perfectly coalesced (512 B / instr / wave)
  auto issue = [&](float* b, int r) {
    const size_t base = (size_t)r * N_SAMPLES;
    async_b128(sigma + base + lane * 4, b +       lane * 4);
    async_b128(dists + base + lane * 4, b + 128 + lane * 4);
    async_b128(zvals + base + lane * 4, b + 256 + lane * 4);
    const float* rg = rgb + (size_t)r * (N_SAMPLES * 3);
    async_b128(rg +        lane * 4, b + 384 + lane * 4);
    async_b128(rg + 128 +  lane * 4, b + 512 + lane * 4);
    async_b128(rg + 256 +  lane * 4, b + 640 + lane * 4);
  };

  int ray = wave;
  if (ray >= n_rays) return;

  issue(buf0, ray);
  int cur = 0;

  while (true) {
    const int nxt = ray + n_waves;
    float* cbuf = cur ? buf1 : buf0;
    if (nxt < n_rays) {                  // prefetch next ray into the other buffer
      issue(cur ? buf0 : buf1, nxt);
      wait_asynccnt<6>();                // in-order completion -> current buf ready
    } else {
      wait_asynccnt<0>();
    }

    // ---- load this lane's 4 samples from LDS (b128 reads, conflict-free) ----
    const float4 sg = *(const float4*)(cbuf +       lane * 4);
    const float4 dd = *(const float4*)(cbuf + 128 + lane * 4);
    const float4 zz = *(const float4*)(cbuf + 256 + lane * 4);
    const float4 c0 = *(const float4*)(cbuf + 384 + lane * 12);
    const float4 c1 = *(const float4*)(cbuf + 388 + lane * 12);
    const float4 c2 = *(const float4*)(cbuf + 392 + lane * 12);

    const float s4[4]  = {sg.x, sg.y, sg.z, sg.w};
    const float d4[4]  = {dd.x, dd.y, dd.z, dd.w};
    const float z4[4]  = {zz.x, zz.y, zz.z, zz.w};
    const float cc[12] = {c0.x, c0.y, c0.z, c0.w,
                          c1.x, c1.y, c1.z, c1.w,
                          c2.x, c2.y, c2.z, c2.w};

    // alpha = 1 - exp(-relu(sigma)*dist);  scan factor f = exp(-tau) + 1e-10
    float a[4], f[4];
#pragma unroll
    for (int i = 0; i < 4; ++i) {
      const float tau = fmaxf(s4[i], 0.0f) * d4[i];
      const float e   = __expf(-tau);
      a[i] = 1.0f - e;
      f[i] = e + 1e-10f;
    }

    // cross-lane inclusive product scan of per-lane factor product (wave32)
    float incl = (f[0] * f[1]) * (f[2] * f[3]);
#pragma unroll
    for (int dl = 1; dl < 32; dl <<= 1) {
      const float t = __shfl_up(incl, dl, 32);
      if (lane >= dl) incl *= t;
    }
    float T = __shfl_up(incl, 1, 32);            // exclusive prefix transmittance
    if (lane == 0) T = 1.0f;
    const float Ttot = __shfl(incl, 31, 32);     // no_hit = T after all samples

    // per-sample weights + weighted accumulation (samples 4*lane .. 4*lane+3)
    float cr = 0.f, cg = 0.f, cb = 0.f, cd = 0.f;
#pragma unroll
    for (int i = 0; i < 4; ++i) {
      const float w = a[i] * T;                  // alpha_i * T_before_i
      cr += w * cc[3 * i + 0];
      cg += w * cc[3 * i + 1];
      cb += w * cc[3 * i + 2];
      cd += w * z4[i];
      T *= f[i];
    }

    // butterfly sum across the wave
#pragma unroll
    for (int m = 16; m > 0; m >>= 1) {
      cr += __shfl_xor(cr, m, 32);
      cg += __shfl_xor(cg, m, 32);
      cb += __shfl_xor(cb, m, 32);
      cd += __shfl_xor(cd, m, 32);
    }

    if (lane == 0) {
      const float* b = bg + (size_t)ray * 3;
      float4 o;
      o.x = cr + Ttot * b[0];
      o.y = cg + Ttot * b[1];
      o.z = cb + Ttot * b[2];
      o.w = cd;
      *(float4*)(out + (size_t)ray * 4) = o;
    }

    if (nxt >= n_rays) break;
    ray = nxt;
    cur ^= 1;
  }
}

extern "C" void kernel_launch(void* const* d_in, const int* in_sizes, int n_in,
                              void* d_out, int out_size, void* d_ws, size_t ws_size,
                              hipStream_t stream) {
  (void)n_in; (void)out_size; (void)d_ws; (void)ws_size;
  const float* sigma = (const float*)d_in[0];
  const float* rgb   = (const float*)d_in[1];
  const float* dists = (const float*)d_in[2];
  const float* zv    = (const float*)d_in[3];
  const float* bg    = (const float*)d_in[4];
  float* out = (float*)d_out;

  const int n_rays = in_sizes[0] / N_SAMPLES;            // 65536
  int waves  = (n_rays + RAYS_PER_WAVE - 1) / RAYS_PER_WAVE;
  int blocks = (waves + WAVES_PER_BLOCK - 1) / WAVES_PER_BLOCK;   // 2048
  if (blocks < 1) blocks = 1;

  raymarch_composite_kernel<<<blocks, BLOCK_SIZE, 0, stream>>>(
      sigma, rgb, dists, zv, bg, out, n_rays);
}